// MonarchFFT_43843026157874
// MI455X (gfx1250) — compile-verified
//
#include <hip/hip_runtime.h>

// Monarch FFT, n = 4096 = 64 x 64, rows = 8*512.
// One workgroup (16 wave32) per row; f32 WMMA 16x16x4 for all six real
// 64x64x64 matmuls (2 for the real-input stage, 4 for the complex stage).

typedef __attribute__((ext_vector_type(2))) float v2f;
typedef __attribute__((ext_vector_type(8))) float v8f;

#define PI_F 3.14159265358979323846f

__device__ __forceinline__ v8f wmma_f32(v2f a, v2f b, v8f c) {
    return __builtin_amdgcn_wmma_f32_16x16x4_f32(
        /*neg_a=*/false, a, /*neg_b=*/false, b,
        /*c_mod=*/(short)0, c, /*reuse_a=*/false, /*reuse_b=*/false);
}

extern "C" __global__ __launch_bounds__(512, 1)
void monarch_fft4096_kernel(const float* __restrict__ x,
                            float* __restrict__ out, int rows)
{
    // 64 KB static LDS, 4 x 4096 floats.
    __shared__ float smem[16384];
    float* sX   = smem;            // x row (reused as Re(C) after stage 1)
    float* sFre = smem + 4096;     // cos table, 64-pt DFT (symmetric)
    float* sFim = smem + 8192;     // sin table
    float* sCre = smem;            // Re(C), aliases sX
    float* sCim = smem + 12288;    // Im(C)

    const int row = blockIdx.x;
    const int tid = threadIdx.x;

    // ---- per-block init: DFT table + x row ----
    for (int i = tid; i < 4096; i += 512) {
        const int j = i >> 6, k = i & 63;
        const float ang = (-2.0f * PI_F / 64.0f) * (float)((j * k) & 63);
        float s, c;
        __sincosf(ang, &s, &c);
        sFre[i] = c;
        sFim[i] = s;
    }
    {
        const float4* xin = (const float4*)(x + (size_t)row * 4096);
        float4* sx4 = (float4*)sX;
        for (int i = tid; i < 1024; i += 512) sx4[i] = xin[i];
    }
    __syncthreads();

    const int lane = tid & 31;
    const int wave = tid >> 5;
    const int tm   = (wave >> 2) << 4;   // tile row base (M)
    const int tn   = (wave & 3)  << 4;   // tile col base (N)
    const int lm   = lane & 15;
    const int koff = (lane >> 4) << 1;   // A/B frag: K = kk + {0,2} by half-wave
    const int moff = (lane >> 4) << 3;   // C/D frag: M = r + {0,8} by half-wave

    // ---- Stage 1: B = A @ F^T  (A real: x[q*64+p] is column-major A) ----
    v8f bre = {}, bim = {};
#pragma unroll
    for (int kk = 0; kk < 64; kk += 4) {
        const int q0 = kk + koff;
        v2f a, fr, fi;
        a.x  = sX[q0 * 64 + tm + lm];          // A[p][q0]  (p = tm+lm)
        a.y  = sX[(q0 + 1) * 64 + tm + lm];
        fr.x = sFre[q0 * 64 + tn + lm];        // F^T[q][k] = F[q][k] (symmetric)
        fr.y = sFre[(q0 + 1) * 64 + tn + lm];
        fi.x = sFim[q0 * 64 + tn + lm];
        fi.y = sFim[(q0 + 1) * 64 + tn + lm];
        bre = wmma_f32(a, fr, bre);
        bim = wmma_f32(a, fi, bim);
    }
    __syncthreads();   // all waves done reading sX before it becomes Re(C)

    // ---- Twiddle: C[p,k] = B[p,k] * exp(-2*pi*i*p*k/4096), write to LDS ----
#pragma unroll
    for (int r = 0; r < 8; ++r) {
        const int M = tm + r + moff;
        const int N = tn + lm;
        const float ang = (-2.0f * PI_F / 4096.0f) * (float)((M * N) & 4095);
        float s, c;
        __sincosf(ang, &s, &c);
        const float br = bre[r], bi = bim[r];
        sCre[M * 64 + N] = br * c - bi * s;
        sCim[M * 64 + N] = br * s + bi * c;
    }
    __syncthreads();

    // ---- Stage 2: D = F @ C  (complex x complex, 4 real matmuls) ----
    v8f dre = {}, dim = {};
#pragma unroll
    for (int kk = 0; kk < 64; kk += 4) {
        const int k0 = kk + koff;
        v2f are, aim, ain, cr, ci;
        // F[p][k] read as F[k][p] via symmetry -> stride-1 across lanes.
        are.x = sFre[k0 * 64 + tm + lm];
        are.y = sFre[(k0 + 1) * 64 + tm + lm];
        aim.x = sFim[k0 * 64 + tm + lm];
        aim.y = sFim[(k0 + 1) * 64 + tm + lm];
        ain.x = -aim.x;                        // f32 WMMA has no A-neg modifier
        ain.y = -aim.y;
        cr.x  = sCre[k0 * 64 + tn + lm];
        cr.y  = sCre[(k0 + 1) * 64 + tn + lm];
        ci.x  = sCim[k0 * 64 + tn + lm];
        ci.y  = sCim[(k0 + 1) * 64 + tn + lm];
        dre = wmma_f32(are, cr, dre);
        dre = wmma_f32(ain, ci, dre);
        dim = wmma_f32(are, ci, dim);
        dim = wmma_f32(aim, cr, dim);
    }

    // ---- Store D (interleaved complex64) ----
    float2* o = (float2*)out + (size_t)row * 4096;
#pragma unroll
    for (int r = 0; r < 8; ++r) {
        const int M = tm + r + moff;
        const int N = tn + lm;
        float2 v;
        v.x = dre[r];
        v.y = dim[r];
        o[M * 64 + N] = v;    // global_store_b64, coalesced per half-wave
    }
}

extern "C" void kernel_launch(void* const* d_in, const int* in_sizes, int n_in,
                              void* d_out, int out_size, void* d_ws, size_t ws_size,
                              hipStream_t stream) {
    (void)n_in; (void)out_size; (void)d_ws; (void)ws_size;
    const float* x = (const float*)d_in[0];
    float* out = (float*)d_out;
    const int rows = in_sizes[0] / 4096;   // 8*512 = 4096 independent FFTs
    monarch_fft4096_kernel<<<dim3(rows), dim3(512), 0, stream>>>(x, out, rows);
}